// dogtaitujuanji_8349416423576
// MI455X (gfx1250) — compile-verified
//
#include <hip/hip_runtime.h>
#include <math.h>

typedef __attribute__((ext_vector_type(2))) float v2f;
typedef __attribute__((ext_vector_type(4))) float v4f;
typedef __attribute__((ext_vector_type(8))) float v8f;
typedef unsigned int u32x4 __attribute__((ext_vector_type(4)));
typedef int          i32x4 __attribute__((ext_vector_type(4)));
typedef int          i32x8 __attribute__((ext_vector_type(8)));

#define NUMN   65536
#define NFEAT  768
#define KCAT   512
#define BN_EPS 1e-5f

// ---------------------------------------------------------------------------
// Tensor Data Mover: async 2D tile load Global -> LDS (per ISA 08, sec 7/8).
// Builds D# group0 (count/lds_addr/global_addr/type=2) and group1
// (data_size=4B, tensor_dim0/1, tile_dim0/1, dim0_stride, optional LDS pad).
// Tracked by TENSORcnt; caller waits with s_wait_tensorcnt.
// ---------------------------------------------------------------------------
__device__ __forceinline__ void tdm_load_2d(
    unsigned lds_byte_off, const void* gptr,
    unsigned tensor_d0, unsigned tensor_d1, unsigned d0_stride,
    unsigned tile_d0, unsigned tile_d1,
    unsigned pad_interval_code, unsigned pad_amount_code, bool pad_en)
{
    const unsigned long long ga = (unsigned long long)gptr;
    u32x4 g0;
    g0[0] = 1u;                                        // count=1, user mode
    g0[1] = lds_byte_off;                              // LDS dest (bytes)
    g0[2] = (unsigned)(ga & 0xFFFFFFFFu);              // global_addr[31:0]
    g0[3] = (unsigned)((ga >> 32) & 0x1FFFFFFu)        // global_addr[56:32]
          | (2u << 30);                                // type = 2 ("image")
    i32x8 g1;
    g1[0] = (int)((2u << 16)                           // data_size = 4 bytes
          | (pad_en ? (1u << 20) : 0u)                 // pad_enable
          | (pad_interval_code << 22)                  // 2^(n+1) dwords/row
          | (pad_amount_code  << 25));                 // (n+1) dwords pad
    g1[1] = (int)(tensor_d0 << 16);                    // tensor_dim0[15:0]
    g1[2] = (int)((tensor_d0 >> 16) | (tensor_d1 << 16));
    g1[3] = (int)((tensor_d1 >> 16) | (tile_d0  << 16));
    g1[4] = (int)(tile_d1 & 0xFFFFu);                  // tile_dim1, tile_dim2=0
    g1[5] = (int)d0_stride;                            // dim0_stride[31:0]
    g1[6] = 0;                                         // stride hi / dim1_stride
    g1[7] = 0;
    i32x4 z4 = (i32x4)0;
#if defined(__clang_major__) && (__clang_major__ >= 23)
    i32x8 z8 = (i32x8)0;
    __builtin_amdgcn_tensor_load_to_lds(g0, g1, z4, z4, z8, 0);
#else
    __builtin_amdgcn_tensor_load_to_lds(g0, g1, z4, z4, 0);
#endif
}

// ---------------------------------------------------------------------------
// K1: fuse the 4 weight matrices into Wcat (512 x 768), zero the stats buf.
//   Wa = w2a.w2b + 0.5*w1a.w1b ;  Wb = 0.5*(w1a.w2b + w2a.w1b) + 0.25*w1a.w1b
// ---------------------------------------------------------------------------
__global__ __launch_bounds__(256) void prep_weights(
    const float* __restrict__ w1a, const float* __restrict__ w2a,
    const float* __restrict__ w1b, const float* __restrict__ w2b,
    float* __restrict__ Wcat, float* __restrict__ stats)
{
    const int n = blockIdx.x * 256 + threadIdx.x;   // 0..767
    const int k = blockIdx.y;                       // 0..511
    if (blockIdx.x == 0 && k < 6)
        stats[k * 256 + threadIdx.x] = 0.0f;

    const int kr = (k < 256) ? k : (k - 256);
    const float* r1 = w1a + (long)kr * 768;
    const float* r2 = w2a + (long)kr * 768;
    float d11 = 0.f, d22 = 0.f, d12 = 0.f, d21 = 0.f;
    for (int j = 0; j < 768; ++j) {
        const float a1 = r1[j], a2 = r2[j];
        const float b1 = w1b[(long)j * 768 + n];
        const float b2 = w2b[(long)j * 768 + n];
        d11 = fmaf(a1, b1, d11);
        d22 = fmaf(a2, b2, d22);
        d12 = fmaf(a1, b2, d12);
        d21 = fmaf(a2, b1, d21);
    }
    const float wv = (k < 256) ? (d22 + 0.5f * d11)
                               : (0.5f * (d12 + d21) + 0.25f * d11);
    Wcat[(long)k * 768 + n] = wv;
}

// ---------------------------------------------------------------------------
// K2: out_pre = [x0 | x1+x2] @ Wcat via V_WMMA_F32_16X16X4_F32.
// grid (4, 512): 192N x 128M WG tiles, 8 waves (4M x 2N), wave tile 32x96.
// K=512 in 16 chunks of 32, double-buffered LDS.
//  - B chunk (32x192 of Wcat): TDM async load every chunk.
//  - A chunk: TDM (with HW pad -> stride 36) for the x0 half (k<256),
//             manual x1+x2 staging for the second half.
// Chunk kc+1's TDM is issued before computing chunk kc (latency hiding);
// the issuing wave drains with s_wait_tensorcnt before the publish barrier.
// ---------------------------------------------------------------------------
__global__ __launch_bounds__(256) void gemm_fused(
    const float* __restrict__ x, const float* __restrict__ Wcat,
    float* __restrict__ out)
{
    __shared__ float Alds[2][128 * 36];   // stride 36: 16B aligned, bank-spread
    __shared__ float Blds[2][32 * 192];   // K-major

    const int tid   = threadIdx.x;
    const int mblk  = blockIdx.y;          // 0..511
    const int nblk  = blockIdx.x;          // 0..3
    const int lane  = tid & 31;
    const int wv    = tid >> 5;            // 0..7
    const int rm    = wv >> 1;             // 0..3 : 32-row slice
    const int cn    = wv & 1;              // 0..1 : 96-col slice
    const int lm    = lane & 15;
    const int khalf = (lane >> 4) * 2;     // k-pair select (ISA A/B layout)

    v8f acc[2][6];
    #pragma unroll
    for (int t = 0; t < 2; ++t)
        #pragma unroll
        for (int j = 0; j < 6; ++j) acc[t][j] = (v8f){};

    const long rowbase = (long)mblk * 128;
    const unsigned aoff0 = (unsigned)(unsigned long long)&Alds[0][0];
    const unsigned aoff1 = (unsigned)(unsigned long long)&Alds[1][0];
    const unsigned boff0 = (unsigned)(unsigned long long)&Blds[0][0];
    const unsigned boff1 = (unsigned)(unsigned long long)&Blds[1][0];

    // ---- prologue: async-load chunk 0 into buffer 0 ----
    if (wv == 0) {
        tdm_load_2d(boff0, Wcat + nblk * 192, 768, 512, 768, 192, 32, 0, 0, false);
        tdm_load_2d(aoff0, x + rowbase * 768, 768, 65536, 768, 32, 128, 4, 3, true);
    }

    for (int kc = 0; kc < 16; ++kc) {
        const int b     = kc & 1;
        const int kbase = kc * 32;

        __syncthreads();   // previous compute done -> buffer 1-b reusable

        // ---- issue next chunk's TDM into the other buffer ----
        if (wv == 0 && kc < 15) {
            const int kn = kbase + 32;
            const unsigned ao = b ? aoff0 : aoff1;
            const unsigned bo = b ? boff0 : boff1;
            tdm_load_2d(bo, Wcat + (long)kn * 768 + nblk * 192,
                        768, 512, 768, 192, 32, 0, 0, false);
            if (kn < 256)
                tdm_load_2d(ao, x + rowbase * 768 + kn,
                            768, 65536, 768, 32, 128, 4, 3, true);
        }

        // ---- manual A staging (x1+x2) for the second K half ----
        if (kbase >= 256) {
            #pragma unroll
            for (int i = 0; i < 4; ++i) {
                const int slot = tid + i * 256;        // 0..1023 float4 slots
                const int r    = slot >> 3;            // 0..127
                const int c4   = (slot & 7) * 4;       // 0,4,...,28
                const float* src = x + (rowbase + r) * 768;
                const int ks = (kbase - 256) + c4;
                v4f v1 = *(const v4f*)(src + 256 + ks);
                v4f v2 = *(const v4f*)(src + 512 + ks);
                *(v4f*)(&Alds[b][r * 36 + c4]) = v1 + v2;
            }
        }

        // ---- drain this chunk's TDM (next chunk's may stay in flight) ----
        if (wv == 0) {
            if (kc >= 15)     __builtin_amdgcn_s_wait_tensorcnt(0);
            else if (kc >= 7) __builtin_amdgcn_s_wait_tensorcnt(1);
            else              __builtin_amdgcn_s_wait_tensorcnt(2);
        }
        __syncthreads();   // publish A+B of chunk kc to all waves

        // ---- 8 WMMA K-steps of 4 ----
        #pragma unroll
        for (int ks = 0; ks < 32; ks += 4) {
            const int k0 = ks + khalf;
            v2f a0 = *(const v2f*)(&Alds[b][(rm * 32 +  0 + lm) * 36 + k0]);
            v2f a1 = *(const v2f*)(&Alds[b][(rm * 32 + 16 + lm) * 36 + k0]);
            #pragma unroll
            for (int j = 0; j < 6; ++j) {
                const int n = cn * 96 + j * 16 + lm;
                v2f bb;
                bb.x = Blds[b][k0 * 192 + n];
                bb.y = Blds[b][(k0 + 1) * 192 + n];
                acc[0][j] = __builtin_amdgcn_wmma_f32_16x16x4_f32(
                    false, a0, false, bb, (short)0, acc[0][j], false, false);
                acc[1][j] = __builtin_amdgcn_wmma_f32_16x16x4_f32(
                    false, a1, false, bb, (short)0, acc[1][j], false, false);
            }
        }
    }

    // ---- store: C layout VGPR r -> row r (lanes 0-15) / r+8 (lanes 16-31) ----
    const int mwave = mblk * 128 + rm * 32;
    const int nwave = nblk * 192 + cn * 96;
    const int mhalf = (lane >> 4) << 3;
    #pragma unroll
    for (int t = 0; t < 2; ++t) {
        const int m0 = mwave + t * 16 + mhalf;
        #pragma unroll
        for (int j = 0; j < 6; ++j) {
            const int col = nwave + j * 16 + lm;
            #pragma unroll
            for (int r = 0; r < 8; ++r)
                out[(long)(m0 + r) * 768 + col] = acc[t][j][r];
        }
    }
}

// ---------------------------------------------------------------------------
// K3: per-feature column sums / sum-of-squares over 65536 rows.
// ---------------------------------------------------------------------------
__global__ __launch_bounds__(256) void colstats(
    const float* __restrict__ out, float* __restrict__ stats)
{
    const int col = blockIdx.x * 256 + threadIdx.x;
    const long r0 = (long)blockIdx.y * 1024;
    float s = 0.f, s2 = 0.f;
    for (int i = 0; i < 1024; ++i) {
        const float v = out[(r0 + i) * 768 + col];
        s  += v;
        s2 = fmaf(v, v, s2);
    }
    atomicAdd(&stats[col], s);
    atomicAdd(&stats[768 + col], s2);
}

// ---------------------------------------------------------------------------
// K4: in-place batchnorm + sigmoid. One float4 per thread.
// ---------------------------------------------------------------------------
__global__ __launch_bounds__(256) void bn_sigmoid(
    float* __restrict__ out, const float* __restrict__ stats,
    const float* __restrict__ gamma, const float* __restrict__ beta)
{
    const long base = ((long)blockIdx.x * 256 + threadIdx.x) * 4;
    const int  col  = (int)(base % 768);
    v4f v = *(const v4f*)(out + base);
    v4f o;
    #pragma unroll
    for (int q = 0; q < 4; ++q) {
        const int   c    = col + q;
        const float mean = stats[c] * (1.0f / 65536.0f);
        const float var  = stats[768 + c] * (1.0f / 65536.0f) - mean * mean;
        const float z    = (v[q] - mean) * rsqrtf(var + BN_EPS) * gamma[c] + beta[c];
        o[q] = 1.0f / (1.0f + __expf(-z));
    }
    *(v4f*)(out + base) = o;
}

// ---------------------------------------------------------------------------
extern "C" void kernel_launch(void* const* d_in, const int* in_sizes, int n_in,
                              void* d_out, int out_size, void* d_ws, size_t ws_size,
                              hipStream_t stream)
{
    const float* x     = (const float*)d_in[0];
    const float* w1a   = (const float*)d_in[1];
    const float* w2a   = (const float*)d_in[2];
    const float* w1b   = (const float*)d_in[3];
    const float* w2b   = (const float*)d_in[4];
    const float* gamma = (const float*)d_in[5];
    const float* beta  = (const float*)d_in[6];
    float* out   = (float*)d_out;
    float* Wcat  = (float*)d_ws;                 // 512*768 floats = 1.5 MB
    float* stats = Wcat + (long)KCAT * NFEAT;    // 1536 floats (sums | sumsq)

    prep_weights<<<dim3(3, 512), 256, 0, stream>>>(w1a, w2a, w1b, w2b, Wcat, stats);
    gemm_fused  <<<dim3(4, 512), 256, 0, stream>>>(x, Wcat, out);
    colstats    <<<dim3(3, 64),  256, 0, stream>>>(out, stats);
    bn_sigmoid  <<<49152,        256, 0, stream>>>(out, stats, gamma, beta);
}